// UGCN_28355374088793
// MI455X (gfx1250) — compile-verified
//
#include <hip/hip_runtime.h>
#include <hip/hip_bf16.h>

typedef __attribute__((ext_vector_type(2))) float v2f;
typedef __attribute__((ext_vector_type(8))) float v8f;

// ---------------- degree / normalization ----------------
__global__ void gcn_fill1(float* __restrict__ p, int n) {
  int i = blockIdx.x * blockDim.x + threadIdx.x;
  if (i < n) p[i] = 1.0f;  // self-loop contributes 1 to every degree
}

__global__ void gcn_degcount(const int* __restrict__ col, float* __restrict__ deg, int E) {
  int e = blockIdx.x * blockDim.x + threadIdx.x;
  if (e < E) atomicAdd(&deg[col[e]], 1.0f);
}

__global__ void gcn_rsqrt(float* __restrict__ p, int n) {
  int i = blockIdx.x * blockDim.x + threadIdx.x;
  if (i < n) p[i] = rsqrtf(p[i]);  // deg >= 1 always (self loops)
}

// ---------------- WMMA f32 GEMM: C[N x NC] = A[N x K] @ W[K x NC] ----------------
// One wave computes one 16x16 output tile via 32x V_WMMA_F32_16X16X4_F32.
// 256 threads = 8 waves: ROW_TILES x COL_TILES = 8 tiles per block.
template <int K, int NC, int ROW_TILES>
__launch_bounds__(256)
__global__ void gcn_gemm_wmma(const float* __restrict__ A, const float* __restrict__ W,
                              float* __restrict__ C, int nrows) {
  constexpr int KP = K + 4;          // pad LDS rows: stride 132 words -> 4-bank stagger
  constexpr int COL_TILES = NC / 16;
  static_assert(ROW_TILES * COL_TILES == 8, "8 waves per block");
  __shared__ float sW[K * NC];
  __shared__ float sA[ROW_TILES * 16 * KP];

  const int rowBase = blockIdx.x * (ROW_TILES * 16);

  // Stage W (L2-resident, broadcast to all blocks) into LDS with float4 loads.
  for (int i = threadIdx.x; i < (K * NC) / 4; i += 256)
    ((float4*)sW)[i] = ((const float4*)W)[i];

  // Stage the A row-tile into padded LDS.
  for (int i = threadIdx.x; i < ROW_TILES * 16 * (K / 4); i += 256) {
    int r = i / (K / 4);
    int kk = i % (K / 4);
    int gr = rowBase + r;
    if (gr > nrows - 1) gr = nrows - 1;  // clamp (keeps EXEC full for WMMA later)
    float4 v = ((const float4*)(A + (size_t)gr * K))[kk];
    float* dst = &sA[r * KP + kk * 4];
    dst[0] = v.x; dst[1] = v.y; dst[2] = v.z; dst[3] = v.w;
  }
  __syncthreads();

  const int wave = threadIdx.x >> 5;
  const int lane = threadIdx.x & 31;
  const int rowTile = wave / COL_TILES;
  const int colTile = wave % COL_TILES;
  const int m  = lane & 15;   // A: row within tile | B/D: column within tile
  const int hi = lane >> 4;   // half-wave selector

  const float* aRow  = &sA[(rowTile * 16 + m) * KP];
  const float* bBase = &sW[colTile * 16 + m];

  v8f acc = {};
#pragma unroll
  for (int k0 = 0; k0 < K; k0 += 4) {
    // A 16x4 f32 fragment: v0 = K+0 (lo half) / K+2 (hi half); v1 = K+1 / K+3
    v2f a, b;
    a.x = aRow[k0 + 2 * hi];
    a.y = aRow[k0 + 2 * hi + 1];
    // B 4x16 f32 fragment: same K split across halves, N = lane&15
    b.x = bBase[(k0 + 2 * hi) * NC];
    b.y = bBase[(k0 + 2 * hi + 1) * NC];
    acc = __builtin_amdgcn_wmma_f32_16x16x4_f32(
        /*neg_a=*/false, a, /*neg_b=*/false, b,
        /*c_mod=*/(short)0, acc, /*reuse_a=*/false, /*reuse_b=*/false);
  }

  // D layout: lane half selects M offset {0,8}; VGPR v selects M=v; N = lane&15.
  const int gcol = colTile * 16 + m;
#pragma unroll
  for (int v = 0; v < 8; ++v) {
    int grow = rowBase + rowTile * 16 + v + 8 * hi;
    if (grow < nrows) C[(size_t)grow * NC + gcol] = acc[v];
  }
}

// ---------------- self-loop term: out[i,:] = dinv[i]^2 * h[i,:] ----------------
template <int LOGF>
__global__ void gcn_selfloop(const float* __restrict__ h, const float* __restrict__ dinv,
                             float* __restrict__ out, int total) {
  int idx = blockIdx.x * blockDim.x + threadIdx.x;
  if (idx < total) {
    float d = dinv[idx >> LOGF];
    out[idx] = d * d * h[idx];
  }
}

// ---------------- edge scatter, F=128: one wave per edge ----------------
__global__ void gcn_scatter128(const float* __restrict__ h, const int* __restrict__ row,
                               const int* __restrict__ col, const float* __restrict__ dinv,
                               float* __restrict__ out, int E) {
  int e = (int)(((size_t)blockIdx.x * blockDim.x + threadIdx.x) >> 5);
  if (e >= E) return;
  int lane = threadIdx.x & 31;
  int r = row[e], c = col[e];
  __builtin_prefetch(h + (size_t)r * 128, 0, 1);  // global_prefetch of source row
  float w = dinv[r] * dinv[c];
  float4 v = ((const float4*)(h + (size_t)r * 128))[lane];  // 16B/lane, full row per wave
  float* dst = out + (size_t)c * 128 + lane * 4;
  atomicAdd(dst + 0, w * v.x);
  atomicAdd(dst + 1, w * v.y);
  atomicAdd(dst + 2, w * v.z);
  atomicAdd(dst + 3, w * v.w);
}

// ---------------- edge scatter, F=64: two edges per wave ----------------
__global__ void gcn_scatter64(const float* __restrict__ h, const int* __restrict__ row,
                              const int* __restrict__ col, const float* __restrict__ dinv,
                              float* __restrict__ out, int E) {
  size_t tid = (size_t)blockIdx.x * blockDim.x + threadIdx.x;
  int lane = threadIdx.x & 31;
  int e = (int)((tid >> 5) * 2) + (lane >> 4);
  if (e >= E) return;
  int l16 = lane & 15;
  int r = row[e], c = col[e];
  float w = dinv[r] * dinv[c];
  float4 v = ((const float4*)(h + (size_t)r * 64))[l16];
  float* dst = out + (size_t)c * 64 + l16 * 4;
  atomicAdd(dst + 0, w * v.x);
  atomicAdd(dst + 1, w * v.y);
  atomicAdd(dst + 2, w * v.z);
  atomicAdd(dst + 3, w * v.w);
}

// ---------------- bias + ReLU ----------------
template <int F>
__global__ void gcn_bias_relu(float* __restrict__ a, const float* __restrict__ b, int total) {
  int idx = blockIdx.x * blockDim.x + threadIdx.x;
  if (idx < total) a[idx] = fmaxf(a[idx] + b[idx & (F - 1)], 0.0f);
}

// ---------------- deterministic two-stage column-mean (F=64) ----------------
__global__ void gcn_reduce64(const float* __restrict__ a, float* __restrict__ partials, int n) {
  __shared__ float s[4 * 64];
  int g = threadIdx.x >> 6;     // 4 node-groups per block
  int cmod = threadIdx.x & 63;  // column
  float acc = 0.0f;
  for (int i = blockIdx.x * 4 + g; i < n; i += gridDim.x * 4)
    acc += a[(size_t)i * 64 + cmod];
  s[threadIdx.x] = acc;
  __syncthreads();
  if (g == 0)
    partials[blockIdx.x * 64 + cmod] = s[cmod] + s[64 + cmod] + s[128 + cmod] + s[192 + cmod];
}

__global__ void gcn_final64(const float* __restrict__ partials, const float* __restrict__ b2,
                            float* __restrict__ out, int nb, float invN) {
  int c = threadIdx.x;  // 64 threads
  float acc = 0.0f;
  for (int i = 0; i < nb; ++i) acc += partials[i * 64 + c];
  out[c] = acc * invN + b2[c];
}

// ---------------- launch ----------------
extern "C" void kernel_launch(void* const* d_in, const int* in_sizes, int n_in,
                              void* d_out, int out_size, void* d_ws, size_t ws_size,
                              hipStream_t stream) {
  const float* x  = (const float*)d_in[0];
  const int*   ei = (const int*)d_in[1];   // [2, E] int32
  const float* W1 = (const float*)d_in[2];
  const float* b1 = (const float*)d_in[3];
  const float* W2 = (const float*)d_in[4];
  const float* b2 = (const float*)d_in[5];

  const int N = in_sizes[0] / 128;
  const int E = in_sizes[1] / 2;
  const int* rows = ei;
  const int* cols = ei + E;

  // workspace carve-up (floats): dinv[N] | partials[240*64] | bufA[N*128] | bufB[N*128]
  const int NB_RED = 240;
  float* ws       = (float*)d_ws;
  float* dinv     = ws;
  float* partials = dinv + N;
  float* bufA     = partials + (size_t)NB_RED * 64;
  float* bufB     = bufA + (size_t)N * 128;

  const int TPB = 256;
  // normalization: deg (with self loop) -> dinv = rsqrt(deg)
  gcn_fill1<<<(N + TPB - 1) / TPB, TPB, 0, stream>>>(dinv, N);
  gcn_degcount<<<(E + TPB - 1) / TPB, TPB, 0, stream>>>(cols, dinv, E);
  gcn_rsqrt<<<(N + TPB - 1) / TPB, TPB, 0, stream>>>(dinv, N);

  // ---- layer 1: h0 = x @ W1 (WMMA f32), aggregate, +b1, relu ----
  gcn_gemm_wmma<128, 128, 1><<<(N + 15) / 16, TPB, 0, stream>>>(x, W1, bufA, N);
  int tot1 = N * 128;
  gcn_selfloop<7><<<(tot1 + TPB - 1) / TPB, TPB, 0, stream>>>(bufA, dinv, bufB, tot1);
  {
    size_t blocks = (((size_t)E * 32) + TPB - 1) / TPB;  // one wave per edge
    gcn_scatter128<<<blocks, TPB, 0, stream>>>(bufA, rows, cols, dinv, bufB, E);
  }
  gcn_bias_relu<128><<<(tot1 + TPB - 1) / TPB, TPB, 0, stream>>>(bufB, b1, tot1);

  // ---- layer 2: h2 = h @ W2 (WMMA f32), aggregate, mean + b2 ----
  gcn_gemm_wmma<128, 64, 2><<<(N + 31) / 32, TPB, 0, stream>>>(bufB, W2, bufA, N);
  int tot2 = N * 64;
  gcn_selfloop<6><<<(tot2 + TPB - 1) / TPB, TPB, 0, stream>>>(bufA, dinv, bufB, tot2);
  {
    size_t waves = ((size_t)E + 1) / 2;                  // two edges per wave
    size_t blocks = ((waves * 32) + TPB - 1) / TPB;
    gcn_scatter64<<<blocks, TPB, 0, stream>>>(bufA, rows, cols, dinv, bufB, E);
  }
  gcn_reduce64<<<NB_RED, TPB, 0, stream>>>(bufB, partials, N);
  gcn_final64<<<1, 64, 0, stream>>>(partials, b2, (float*)d_out, NB_RED, 1.0f / (float)N);
}